// BlockwiseLinearFusion_429496730340
// MI455X (gfx1250) — compile-verified
//
#include <hip/hip_runtime.h>

typedef __attribute__((ext_vector_type(2))) float v2f;
typedef __attribute__((ext_vector_type(8))) float v8f;

// Problem constants (match reference)
#define NB  4          // batch
#define NK  32         // K contraction
#define NC  3          // channels
#define NH  512
#define NW  512
#define BS  16         // block size
#define HB  (NH / BS)  // 32
#define WB  (NW / BS)  // 32

// out[b,c,i*16+p, j*16+q] = sum_k x[b,k,c,i*16+p,j*16+q] * w[i,j,c,k]
//
// One wave handles one (b,c,i,j) 16x16 output block.
// Per pixel row p: D(16x16) = sum over 8 chunks of A(16x4: weights bcast) x B(4x16: x pixels)
// accumulated in f32 via v_wmma_f32_16x16x4_f32. Row 0 of D (VGPR0, lanes 0..15) = outputs.
__global__ __launch_bounds__(256)
void blockwise_linear_wmma(const float* __restrict__ x,
                           const float* __restrict__ w,
                           float* __restrict__ out) {
  const int wid  = blockIdx.x * 8 + (threadIdx.x >> 5);   // global wave id
  const int lane = threadIdx.x & 31;
  if (wid >= NB * NC * HB * WB) return;

  // wid -> (b, c, i, j), j fastest
  const int j = wid & (WB - 1);
  int t = wid >> 5;
  const int i = t & (HB - 1);
  t >>= 5;
  const int c = t % NC;
  const int b = t / NC;

  // Per-lane k selector for the 16x4 A / 4x16 B VGPR layouts:
  // lanes 0-15 carry K=0,1 in VGPR0/1; lanes 16-31 carry K=2,3.
  const int ksel = (lane >> 4) << 1;  // 0 or 2

  // ---- A matrix: weights[i][j][c][k], broadcast across the 16 rows ----
  const float* __restrict__ wp = w + (size_t)(((i * WB + j) * NC + c)) * NK;
  v2f A[8];
#pragma unroll
  for (int kb = 0; kb < 8; ++kb) {
    A[kb].x = wp[kb * 4 + ksel];
    A[kb].y = wp[kb * 4 + ksel + 1];
  }

  // ---- x base pointer for this wave's block ----
  const size_t CHW = (size_t)NC * NH * NW;   // stride between consecutive k
  const float* __restrict__ xb =
      x + ((size_t)b * NK + (size_t)ksel) * CHW      // batch + lane-half k offset
        + (size_t)c * NH * NW
        + (size_t)(i * BS) * NW + (size_t)(j * BS)
        + (size_t)(lane & 15);                        // pixel column q

  float* __restrict__ ob =
      out + (size_t)(b * NC + c) * NH * NW
          + (size_t)(i * BS) * NW + (size_t)(j * BS);

#pragma unroll 2
  for (int p = 0; p < BS; ++p) {
    const float* __restrict__ xp = xb + (size_t)p * NW;

    // Gather the 16 B-operand dwords for this pixel row (coalesced 64B halves).
    float b0[8], b1[8];
#pragma unroll
    for (int kb = 0; kb < 8; ++kb) {
      b0[kb] = xp[(size_t)(kb * 4) * CHW];        // k = 4kb + ksel
      b1[kb] = xp[(size_t)(kb * 4 + 1) * CHW];    // k = 4kb + ksel + 1
    }

    v8f acc = {0.f, 0.f, 0.f, 0.f, 0.f, 0.f, 0.f, 0.f};
#pragma unroll
    for (int kb = 0; kb < 8; ++kb) {
      v2f Bv;
      Bv.x = b0[kb];
      Bv.y = b1[kb];
      // 8 args: (neg_a, A, neg_b, B, c_mod, C, reuse_a, reuse_b)
      acc = __builtin_amdgcn_wmma_f32_16x16x4_f32(
          false, A[kb], false, Bv, (short)0, acc, false, false);
    }

    // Every row of D is identical; VGPR0 on lanes 0..15 holds out[q = lane].
    if (lane < 16) {
      ob[(size_t)p * NW + lane] = acc[0];
    }
  }
}

extern "C" void kernel_launch(void* const* d_in, const int* in_sizes, int n_in,
                              void* d_out, int out_size, void* d_ws, size_t ws_size,
                              hipStream_t stream) {
  (void)in_sizes; (void)n_in; (void)d_ws; (void)ws_size; (void)out_size;
  const float* x = (const float*)d_in[0];   // (B,K,C,H,W) f32
  const float* w = (const float*)d_in[1];   // (Hb,Wb,C,K) f32
  float* out = (float*)d_out;               // (B,C,H,W) f32

  const int totalWaves = NB * NC * HB * WB;       // 12288
  const int wavesPerWG = 8;                        // 256 threads, wave32
  const int grid = (totalWaves + wavesPerWG - 1) / wavesPerWG;  // 1536
  blockwise_linear_wmma<<<grid, 256, 0, stream>>>(x, w, out);
}